// AsymmetricContrastiveLoss_30588757082779
// MI455X (gfx1250) — compile-verified
//
#include <hip/hip_runtime.h>
#include <math.h>

// ---------------------------------------------------------------------------
// AsymmetricContrastiveLoss for MI455X (gfx1250, wave32).
// Bandwidth-bound (z = 256 MB fp32). Per-row 4x4 timepoint Grams via
// V_WMMA_F32_16X16X4_F32 (4 rows -> 16x16 tile, diagonal 4x4 blocks).
// Tile staging global->LDS via the Tensor Data Mover (tensor_load_to_lds),
// double-buffered with s_wait_tensorcnt.
// ---------------------------------------------------------------------------

#define EPSF 1e-8f

typedef __attribute__((ext_vector_type(2))) float v2f;
typedef __attribute__((ext_vector_type(4))) float v4f;
typedef __attribute__((ext_vector_type(8))) float v8f;
typedef __attribute__((ext_vector_type(4))) unsigned v4u;
typedef __attribute__((ext_vector_type(4))) int v4i;
typedef __attribute__((ext_vector_type(8))) int v8i;

#if __has_builtin(__builtin_amdgcn_tensor_load_to_lds) && \
    __has_builtin(__builtin_amdgcn_s_wait_tensorcnt)
#define TILE_TDM 1
#elif __has_builtin(__builtin_amdgcn_global_load_async_to_lds_b128) && \
    __has_builtin(__builtin_amdgcn_s_wait_asynccnt)
#define TILE_ASYNC 1
#else
#define TILE_SYNC 1
#endif

#define GRAM_WAVES 4
#define GRAM_BLOCK 128
#define KCH 64
#define LDS_STRIDE 68  // 64 data + 4 pad DWORDs (TDM pad_interval=5, pad_amount=3)

#define RFL(x) ((unsigned)__builtin_amdgcn_readfirstlane((int)(x)))

// ===================== threefry2x32 (exact JAX semantics) ====================
__device__ __forceinline__ void threefry2x32(unsigned k0, unsigned k1,
                                             unsigned x0, unsigned x1,
                                             unsigned& o0, unsigned& o1) {
  unsigned ks[3] = {k0, k1, k0 ^ k1 ^ 0x1BD11BDAu};
  x0 += ks[0];
  x1 += ks[1];
  const int rotA[4] = {13, 15, 26, 6};
  const int rotB[4] = {17, 29, 16, 24};
#pragma unroll
  for (int i = 0; i < 5; ++i) {
    const int* r = (i & 1) ? rotB : rotA;
#pragma unroll
    for (int j = 0; j < 4; ++j) {
      x0 += x1;
      x1 = (x1 << r[j]) | (x1 >> (32 - r[j]));
      x1 ^= x0;
    }
    x0 += ks[(i + 1) % 3];
    x1 += ks[(i + 2) % 3] + (unsigned)(i + 1);
  }
  o0 = x0;
  o1 = x1;
}

__device__ __forceinline__ float bits_to_uniform(unsigned b) {
  return __uint_as_float((b >> 9) | 0x3f800000u) - 1.0f;
}

// ============ Kernel 1: stable compaction of positive indices + Pi ==========
__global__ void scan_kernel(const int* __restrict__ labels, int* __restrict__ posIndex,
                            int* __restrict__ scalars, int B) {
  __shared__ int sh[1024];
  __shared__ int carry_s;
  if (threadIdx.x == 0) carry_s = 0;
  __syncthreads();
  for (int c = 0; c < B; c += 1024) {
    int idx = c + (int)threadIdx.x;
    int v = (idx < B && labels[idx] != 0) ? 1 : 0;
    sh[threadIdx.x] = v;
    __syncthreads();
    for (int off = 1; off < 1024; off <<= 1) {
      int add = (threadIdx.x >= (unsigned)off) ? sh[threadIdx.x - off] : 0;
      __syncthreads();
      sh[threadIdx.x] += add;
      __syncthreads();
    }
    int incl = sh[threadIdx.x];
    int base = carry_s;
    if (v) posIndex[base + incl - 1] = idx;
    int total = sh[1023];
    __syncthreads();
    if (threadIdx.x == 0) carry_s = base + total;
    __syncthreads();
  }
  if (threadIdx.x == 0) scalars[0] = carry_s;
}

// ===== Kernel 2: threefry uniforms -> packed (keyBits<<32 | index) keys =====
__global__ void keygen_kernel(const int* __restrict__ labels,
                              unsigned long long* __restrict__ sortA,
                              unsigned long long* __restrict__ sortB, int B) {
  int i = blockIdx.x * blockDim.x + threadIdx.x;
  int H = B >> 1;
  if (i >= H) return;
  unsigned a0, a1, b0, b1;
  threefry2x32(0u, 1u, 0u, 2u, a0, a1);
  threefry2x32(0u, 1u, 1u, 3u, b0, b1);
  unsigned w0, w1;
  threefry2x32(a0, b0, (unsigned)i, (unsigned)(H + i), w0, w1);
  float u1a = bits_to_uniform(w0), u1b = bits_to_uniform(w1);
  threefry2x32(a1, b1, (unsigned)i, (unsigned)(H + i), w0, w1);
  float u2a = bits_to_uniform(w0), u2b = bits_to_uniform(w1);

  int i0 = i, i1 = H + i;
  bool l0 = labels[i0] != 0, l1 = labels[i1] != 0;
  float ka0 = l0 ? u1a : 2.0f, ka1 = l1 ? u1b : 2.0f;
  float kb0 = l0 ? 2.0f : u2a, kb1 = l1 ? 2.0f : u2b;
  sortA[i0] = ((unsigned long long)__float_as_uint(ka0) << 32) | (unsigned)i0;
  sortA[i1] = ((unsigned long long)__float_as_uint(ka1) << 32) | (unsigned)i1;
  sortB[i0] = ((unsigned long long)__float_as_uint(kb0) << 32) | (unsigned)i0;
  sortB[i1] = ((unsigned long long)__float_as_uint(kb1) << 32) | (unsigned)i1;
}

// =================== Kernel 3: bitonic pass over both arrays ================
__global__ void bitonic_pass(unsigned long long* __restrict__ A0,
                             unsigned long long* __restrict__ A1, int j, int k, int N) {
  int tid = blockIdx.x * blockDim.x + threadIdx.x;
  unsigned long long* arr = A0;
  int i = tid;
  if (tid >= N) {
    arr = A1;
    i = tid - N;
    if (i >= N) return;
  }
  int l = i ^ j;
  if (l > i) {
    unsigned long long a = arr[i], b = arr[l];
    bool asc = (i & k) == 0;
    if ((a > b) == asc) {
      arr[i] = b;
      arr[l] = a;
    }
  }
}

// ================= TDM helper: 2D tile (16 x KCH fp32) -> LDS ===============
#if defined(TILE_TDM)
#if __clang_major__ >= 23
#define TDM_LOAD(g0, g1)                                            \
  __builtin_amdgcn_tensor_load_to_lds((g0), (g1), (v4i){0, 0, 0, 0}, \
                                      (v4i){0, 0, 0, 0},             \
                                      (v8i){0, 0, 0, 0, 0, 0, 0, 0}, 0)
#else
#define TDM_LOAD(g0, g1)                                            \
  __builtin_amdgcn_tensor_load_to_lds((g0), (g1), (v4i){0, 0, 0, 0}, \
                                      (v4i){0, 0, 0, 0}, 0)
#endif

// zg: base of the 16 contiguous length-td vectors; kc: column offset.
// Tensor view: dim0 = td (row length, stride td), 16 rows; tile 16 x KCH.
// TDM padding inserts 4 DWORDs after each 64 -> LDS row stride 68 floats.
__device__ __forceinline__ void tdm_issue_tile(const float* zg, int kc,
                                               unsigned lds_byte, int td) {
  unsigned long long ga = (unsigned long long)(const void*)(zg + kc);
  v4u g0;
  g0[0] = RFL(1u);  // count=1, user mode, no gather
  g0[1] = RFL(lds_byte);
  g0[2] = RFL((unsigned)ga);
  g0[3] = RFL(((unsigned)(ga >> 32) & 0x01ffffffu) | (2u << 30));  // addr hi | type=2
  unsigned dim0 = (unsigned)td;
  v8i g1;
  g1[0] = (int)RFL((2u << 16) | (1u << 20) | (5u << 22) | (3u << 25));
  //        data_size=4B     pad_en      pad every 64DW  pad 4 DW
  g1[1] = (int)RFL((dim0 & 0xffffu) << 16);                          // dim0 lo16
  g1[2] = (int)RFL(((dim0 >> 16) & 0xffffu) | (16u << 16));          // dim0 hi | dim1=16
  g1[3] = (int)RFL((unsigned)KCH << 16);                             // tile_dim0=KCH
  g1[4] = (int)RFL(16u);                                             // tile_dim1=16
  g1[5] = (int)RFL(dim0);                                            // stride0 lo32 = td
  g1[6] = (int)RFL(0u);                                              // stride0 hi | stride1 lo
  g1[7] = (int)RFL(0u);
  TDM_LOAD(g0, g1);
}
#endif

// ============ Kernel 4: per-row 4x4 Gram via V_WMMA_F32_16X16X4_F32 =========
__global__ void gram_kernel(const float* __restrict__ z, const int* __restrict__ labels,
                            float* __restrict__ rownorm, float* __restrict__ ortho_p,
                            float* __restrict__ temp_p, int B, int D) {
  const int td = D >> 2;  // 512
  const int lane = threadIdx.x & 31;
  const int w = threadIdx.x >> 5;
  __shared__ float tile[GRAM_WAVES][2][16 * LDS_STRIDE];

  const int numGroups = B >> 2;
  const int waveCount = gridDim.x * GRAM_WAVES;
  const int m16 = lane & 15;
  const int koff = (lane >> 4) * 2;

  for (int g = blockIdx.x * GRAM_WAVES + w; g < numGroups; g += waveCount) {
    v8f acc = {0.f, 0.f, 0.f, 0.f, 0.f, 0.f, 0.f, 0.f};
    // 16 timepoint vectors of this 4-row group are contiguous at zg, each td long
    const float* zg = z + (size_t)(4 * g) * D;

#if defined(TILE_TDM)
    // generic shared pointer: low 32 bits are the LDS byte address
    unsigned lds0 = (unsigned)(unsigned long long)(const void*)&tile[w][0][0];
    unsigned lds1 = (unsigned)(unsigned long long)(const void*)&tile[w][1][0];
    tdm_issue_tile(zg, 0, lds0, td);
    int cur = 0;
    for (int kc = 0; kc < td; kc += KCH) {
      if (kc + KCH < td) {
        tdm_issue_tile(zg, kc + KCH, (cur ? lds0 : lds1), td);
        __builtin_amdgcn_s_wait_tensorcnt(1);  // current buffer ready (in-order)
      } else {
        __builtin_amdgcn_s_wait_tensorcnt(0);
      }
      asm volatile("" ::: "memory");
      const float* my = &tile[w][cur][0];
#pragma unroll
      for (int kk = 0; kk < KCH; kk += 4) {
        v2f a = *(const v2f*)&my[m16 * LDS_STRIDE + kk + koff];
        acc = __builtin_amdgcn_wmma_f32_16x16x4_f32(false, a, false, a, (short)0,
                                                    acc, false, false);
      }
      asm volatile("" ::: "memory");
      cur ^= 1;
    }
#elif defined(TILE_ASYNC)
    typedef __attribute__((address_space(3))) char lds_char;
    typedef __attribute__((address_space(1))) const char g_char;
    for (int kc = 0; kc < td; kc += KCH) {
      float* my = &tile[w][0][0];
#pragma unroll
      for (int i = 0; i < (16 * KCH / 4) / 32; ++i) {
        int s = lane + 32 * i;
        int m = s >> 4;
        int pos = (s & 15) * 4;
        const float* src = zg + (size_t)m * td + kc + pos;
        float* dst = &my[m * LDS_STRIDE + pos];
        __builtin_amdgcn_global_load_async_to_lds_b128(
            (g_char*)(unsigned long long)(const void*)src,
            (lds_char*)(unsigned long long)(unsigned)(unsigned long long)(void*)dst,
            0, 0);
      }
      __builtin_amdgcn_s_wait_asynccnt(0);
      asm volatile("" ::: "memory");
#pragma unroll
      for (int kk = 0; kk < KCH; kk += 4) {
        v2f a = *(const v2f*)&my[m16 * LDS_STRIDE + kk + koff];
        acc = __builtin_amdgcn_wmma_f32_16x16x4_f32(false, a, false, a, (short)0,
                                                    acc, false, false);
      }
      asm volatile("" ::: "memory");
    }
#else  // TILE_SYNC: two-phase staging (8 outstanding loads, then stores)
    for (int kc = 0; kc < td; kc += KCH) {
      float* my = &tile[w][0][0];
      v4f stage[8];
#pragma unroll
      for (int i = 0; i < 8; ++i) {
        int s = lane + 32 * i;
        stage[i] = *(const v4f*)(zg + (size_t)(s >> 4) * td + kc + (s & 15) * 4);
      }
#pragma unroll
      for (int i = 0; i < 8; ++i) {
        int s = lane + 32 * i;
        *(v4f*)&my[(s >> 4) * LDS_STRIDE + (s & 15) * 4] = stage[i];
      }
      __builtin_amdgcn_wave_barrier();
      asm volatile("" ::: "memory");
#pragma unroll
      for (int kk = 0; kk < KCH; kk += 4) {
        v2f a = *(const v2f*)&my[m16 * LDS_STRIDE + kk + koff];
        acc = __builtin_amdgcn_wmma_f32_16x16x4_f32(false, a, false, a, (short)0,
                                                    acc, false, false);
      }
      asm volatile("" ::: "memory");
      __builtin_amdgcn_wave_barrier();
    }
#endif

    // spill C (16x16 Gram; VGPR i: M=i+8*(lane/16), N=lane%16) to LDS
    float* spill = &tile[w][0][0];
    {
      int Mb = 8 * (lane >> 4);
      int N = lane & 15;
#pragma unroll
      for (int i = 0; i < 8; ++i) spill[(Mb + i) * 16 + N] = acc[i];
    }
    asm volatile("" ::: "memory");
    __builtin_amdgcn_wave_barrier();
    if (lane < 4) {
      int r = lane;
      int row = 4 * g + r;
      float G[4][4];
#pragma unroll
      for (int i = 0; i < 4; ++i)
#pragma unroll
        for (int j = 0; j < 4; ++j) G[i][j] = spill[(4 * r + i) * 16 + (4 * r + j)];
      float n0 = fmaxf(sqrtf(G[0][0]), EPSF);
      float n1 = fmaxf(sqrtf(G[1][1]), EPSF);
      float n2 = fmaxf(sqrtf(G[2][2]), EPSF);
      float n3 = fmaxf(sqrtf(G[3][3]), EPSF);
      float po = (fabsf(G[0][1]) / (n0 * n1) + fabsf(G[0][2]) / (n0 * n2) +
                  fabsf(G[0][3]) / (n0 * n3) + fabsf(G[1][2]) / (n1 * n2) +
                  fabsf(G[1][3]) / (n1 * n3) + fabsf(G[2][3]) / (n2 * n3)) *
                 (1.0f / 6.0f);
      float trace = G[0][0] + G[1][1] + G[2][2] + G[3][3];
      rownorm[row] = fmaxf(sqrtf(trace), EPSF);
      // v = z3 - z0: dot(v,z3)=G33-G03, ||v||^2 = G00 - 2*G03 + G33
      float vd = G[3][3] - G[0][3];
      float vv = fmaxf(G[0][0] - 2.0f * G[0][3] + G[3][3], 0.0f);
      float nv = fmaxf(sqrtf(vv), EPSF);
      float pt = 1.0f - vd / (nv * n3);
      float lab = (labels[row] != 0) ? 1.0f : 0.0f;
      ortho_p[row] = lab * po;
      temp_p[row] = lab * pt;
    }
    asm volatile("" ::: "memory");
    __builtin_amdgcn_wave_barrier();
  }
}

// ============== Kernel 5: gathered pair cosines (wave per slot k) ===========
__global__ void align_kernel(const float* __restrict__ z, const int* __restrict__ posIndex,
                             const unsigned long long* __restrict__ sortA,
                             const unsigned long long* __restrict__ sortB,
                             const float* __restrict__ rownorm,
                             const int* __restrict__ scalars,
                             float* __restrict__ alignpos, float* __restrict__ alignneg,
                             int B, int D) {
  int lane = threadIdx.x & 31;
  int w = threadIdx.x >> 5;
  int k = blockIdx.x * 8 + w;
  if (k >= B) return;
  int Pi = scalars[0];
  int Ni = B - Pi;
  int m = Pi < Ni ? Pi : Ni;
  float resPos = 0.0f, resNeg = 0.0f;
  if (k < Pi) {
    int a = posIndex[k];
    int b = (int)(unsigned)(sortA[k] & 0xffffffffull);
    const float4* pa = (const float4*)(z + (size_t)a * D);
    const float4* pb = (const float4*)(z + (size_t)b * D);
    float s = 0.0f;
#pragma unroll 4
    for (int i = lane; i < D / 4; i += 32) {
      float4 x = pa[i], y = pb[i];
      s += x.x * y.x + x.y * y.y + x.z * y.z + x.w * y.w;
    }
#pragma unroll
    for (int off = 16; off > 0; off >>= 1) s += __shfl_xor(s, off, 32);
    resPos = s / (rownorm[a] * rownorm[b]);
    if (k < m) {
      int c = (int)(unsigned)(sortB[k] & 0xffffffffull);
      const float4* pc = (const float4*)(z + (size_t)c * D);
      float t = 0.0f;
#pragma unroll 4
      for (int i = lane; i < D / 4; i += 32) {
        float4 x = pa[i], y = pc[i];
        t += x.x * y.x + x.y * y.y + x.z * y.z + x.w * y.w;
      }
#pragma unroll
      for (int off = 16; off > 0; off >>= 1) t += __shfl_xor(t, off, 32);
      resNeg = t / (rownorm[a] * rownorm[c]);
    }
  }
  if (lane == 0) {
    alignpos[k] = resPos;
    alignneg[k] = resNeg;
  }
}

// ================ Kernel 6: deterministic fixed-order finalize ==============
__device__ float block_sum(const float* __restrict__ v, int n, float* sh) {
  float s = 0.0f;
  for (int i = threadIdx.x; i < (unsigned)n; i += blockDim.x) s += v[i];
  sh[threadIdx.x] = s;
  __syncthreads();
  for (int off = blockDim.x >> 1; off > 0; off >>= 1) {
    if (threadIdx.x < (unsigned)off) sh[threadIdx.x] += sh[threadIdx.x + off];
    __syncthreads();
  }
  float r = sh[0];
  __syncthreads();
  return r;
}

__global__ void finalize_kernel(const float* __restrict__ ortho_p,
                                const float* __restrict__ temp_p,
                                const float* __restrict__ alignpos,
                                const float* __restrict__ alignneg,
                                const int* __restrict__ scalars,
                                float* __restrict__ out, int B) {
  __shared__ float sh[256];
  float s_ortho = block_sum(ortho_p, B, sh);
  float s_temp = block_sum(temp_p, B, sh);
  float s_pos = block_sum(alignpos, B, sh);
  float s_neg = block_sum(alignneg, B, sh);
  if (threadIdx.x == 0) {
    int Pi = scalars[0];
    int Ni = B - Pi;
    int m = Pi < Ni ? Pi : Ni;
    bool haspos = Pi > 0;
    float Pf = fmaxf((float)Pi, 1.0f);
    float mF = fmaxf((float)m, 1.0f);
    float lap = haspos ? (1.0f - s_pos / Pf) : 0.0f;
    float lan = (m > 0) ? (s_neg / mF) : 0.0f;
    out[0] = lap + lan;
    out[1] = haspos ? (s_ortho / Pf) : 0.0f;
    out[2] = haspos ? (s_temp / Pf) : 0.0f;
  }
}

// ============================== launch =====================================
extern "C" void kernel_launch(void* const* d_in, const int* in_sizes, int n_in,
                              void* d_out, int out_size, void* d_ws, size_t ws_size,
                              hipStream_t stream) {
  const float* z = (const float*)d_in[0];
  const int* labels = (const int*)d_in[1];
  const int B = in_sizes[1];      // 32768 (power of two; bitonic assumes it)
  const int D = in_sizes[0] / B;  // 2048

  char* ws = (char*)d_ws;
  size_t off = 0;
  auto carve = [&](size_t bytes) -> void* {
    void* p = ws + off;
    off = (off + bytes + 255) & ~(size_t)255;
    return p;
  };
  int* scalars = (int*)carve(16);
  int* posIndex = (int*)carve(sizeof(int) * B);
  float* rownorm = (float*)carve(sizeof(float) * B);
  unsigned long long* sortA = (unsigned long long*)carve(8ull * B);
  unsigned long long* sortB = (unsigned long long*)carve(8ull * B);
  float* ortho_p = (float*)carve(sizeof(float) * B);
  float* temp_p = (float*)carve(sizeof(float) * B);
  float* alignpos = (float*)carve(sizeof(float) * B);
  float* alignneg = (float*)carve(sizeof(float) * B);

  scan_kernel<<<1, 1024, 0, stream>>>(labels, posIndex, scalars, B);
  keygen_kernel<<<(B / 2 + 255) / 256, 256, 0, stream>>>(labels, sortA, sortB, B);
  for (int k = 2; k <= B; k <<= 1)
    for (int j = k >> 1; j > 0; j >>= 1)
      bitonic_pass<<<(2 * B + 255) / 256, 256, 0, stream>>>(sortA, sortB, j, k, B);
  gram_kernel<<<512, GRAM_BLOCK, 0, stream>>>(z, labels, rownorm, ortho_p, temp_p, B, D);
  align_kernel<<<(B + 7) / 8, 256, 0, stream>>>(z, posIndex, sortA, sortB, rownorm,
                                                scalars, alignpos, alignneg, B, D);
  finalize_kernel<<<1, 256, 0, stream>>>(ortho_p, temp_p, alignpos, alignneg, scalars,
                                         (float*)d_out, B);
}